// SIDARTHEOde_54941221650604
// MI455X (gfx1250) — compile-verified
//
#include <hip/hip_runtime.h>

// SIDARTHE RK4 (3/8 rule), 999 sequential steps, 8-state f32 system.
// Engine: single-lane scalar-pipe FMA chain (latency-optimal; no batch axis,
// so WMMA cannot shorten the dependence chain and is deliberately not used).
// Data path (gfx1250-specific): trajectory staged in LDS via the DS pipe,
// then bulk written back HBM-side with the async LDS->global mover
// (global_store_async_from_lds_b128, ASYNCcnt), 512 B per wave-instruction.

namespace {

struct SidC {
  float alpha, beta, gamma, delta;                       // force coefficients
  float eps, zeta, eta, theta, mu, nu, tau, lam, rho, kappa, xi, sigma;
  float ncI, ncD, ncA, ncR, ncT;                         // negated decay sums
};

__device__ __forceinline__ void sid_rhs(const float y[8], float k[8], const SidC& c) {
  // force = S * (alpha*I + beta*D + gamma*A + delta*R)
  const float force = y[0] * fmaf(c.alpha, y[1],
                            fmaf(c.beta,  y[2],
                            fmaf(c.gamma, y[3], c.delta * y[4])));
  k[0] = force;                                                   // dS (sign as in source)
  k[1] = fmaf(c.ncI,  y[1], force);                               // dI
  k[2] = fmaf(c.eps,  y[1], c.ncD * y[2]);                        // dD
  k[3] = fmaf(c.zeta, y[1], c.ncA * y[3]);                        // dA
  k[4] = fmaf(c.eta,  y[2], fmaf(c.theta, y[3], c.ncR * y[4]));   // dR
  k[5] = fmaf(c.mu,   y[3], fmaf(c.nu,    y[4], c.ncT * y[5]));   // dT
  k[6] = fmaf(c.lam,  y[1],
         fmaf(c.rho,  y[2],
         fmaf(c.kappa,y[3],
         fmaf(c.xi,   y[4], c.sigma * y[5]))));                   // dH
  k[7] = c.tau * y[5];                                            // dE
}

__global__ __launch_bounds__(32, 1)
void sidarthe_rk4_kernel(const float* __restrict__ y0,
                         const float* __restrict__ rates,
                         const int*   __restrict__ steps,
                         float*       __restrict__ out) {
  // Whole 1000x8 f32 trajectory staged in LDS (32 KB of the WGP's 320 KB).
  __shared__ __align__(16) float traj[1000 * 8];

  const unsigned lane = threadIdx.x;

  if (lane == 0) {
    // rates order: alpha,beta,delta,gamma,eps,theta,zeta,eta,mu,nu,tau,lam,rho,kappa,xi,sigma
    SidC c;
    c.alpha = rates[0];  c.beta  = rates[1];
    c.delta = rates[2];  c.gamma = rates[3];
    c.eps   = rates[4];  c.theta = rates[5];
    c.zeta  = rates[6];  c.eta   = rates[7];
    c.mu    = rates[8];  c.nu    = rates[9];
    c.tau   = rates[10]; c.lam   = rates[11];
    c.rho   = rates[12]; c.kappa = rates[13];
    c.xi    = rates[14]; c.sigma = rates[15];
    c.ncI = -(c.eps + c.zeta + c.lam);
    c.ncD = -(c.eta + c.rho);
    c.ncA = -(c.theta + c.mu + c.kappa);
    c.ncR = -(c.nu + c.xi);
    c.ncT = -(c.sigma + c.tau);

    const float dt    = (float)steps[0] / 999.0f;   // N_POINTS-1
    const float third = 1.0f / 3.0f;
    const float dt8   = dt * 0.125f;

    float y[8];
#pragma unroll
    for (int i = 0; i < 8; ++i) y[i] = y0[i];

    // row 0 = y0 (DS pipe, never blocks the chain)
    reinterpret_cast<float4*>(traj)[0] = make_float4(y[0], y[1], y[2], y[3]);
    reinterpret_cast<float4*>(traj)[1] = make_float4(y[4], y[5], y[6], y[7]);

#pragma unroll 1
    for (int t = 1; t < 1000; ++t) {
      float k1[8], k2[8], k3[8], k4[8], yt[8], e1[8];

      sid_rhs(y, k1, c);
#pragma unroll
      for (int i = 0; i < 8; ++i) {               // y + (dt*k1)*third
        e1[i] = dt * k1[i];
        yt[i] = fmaf(third, e1[i], y[i]);
      }
      sid_rhs(yt, k2, c);
#pragma unroll
      for (int i = 0; i < 8; ++i) {               // y + dt*(k2 - k1*third)
        const float m = fmaf(-third, k1[i], k2[i]);
        yt[i] = fmaf(dt, m, y[i]);
      }
      sid_rhs(yt, k3, c);
#pragma unroll
      for (int i = 0; i < 8; ++i) {               // y + dt*(k1 - k2 + k3)
        const float m = (k1[i] - k2[i]) + k3[i];
        yt[i] = fmaf(dt, m, y[i]);
      }
      sid_rhs(yt, k4, c);
#pragma unroll
      for (int i = 0; i < 8; ++i) {               // y += dt*(k1 + 3*(k2+k3) + k4)/8
        const float u = fmaf(3.0f, k2[i] + k3[i], k1[i] + k4[i]);
        y[i] = fmaf(dt8, u, y[i]);
      }

      float* row = traj + (size_t)t * 8;          // 16B-aligned LDS row
      reinterpret_cast<float4*>(row)[0] = make_float4(y[0], y[1], y[2], y[3]);
      reinterpret_cast<float4*>(row)[1] = make_float4(y[4], y[5], y[6], y[7]);
    }
  }

  // Same wave re-converges here (single wave32 workgroup -> no barrier needed),
  // but all staging DS stores must land before the async engine reads LDS.
  asm volatile("s_wait_dscnt 0x0" ::: "memory");

  // Bulk async write-back LDS -> HBM: 2000 float4 chunks, full wave.
  // One wave-instruction moves 32 lanes x 16 B = 512 B, tracked by ASYNCcnt.
  for (unsigned i = lane; i < 2000u; i += 32u) {
    unsigned long long gaddr =
        (unsigned long long)(uintptr_t)out + (unsigned long long)i * 16ull;
    // Generic LDS address: hardware uses addr[31:0] as the DS-unit address.
    unsigned laddr = (unsigned)(uintptr_t)(traj + (size_t)i * 4u);
    asm volatile("global_store_async_from_lds_b128 %0, %1, off"
                 :: "v"(gaddr), "v"(laddr)
                 : "memory");
  }

  // Drain the async mover before kernel end (also covered by the implicit
  // wait-idle at s_endpgm; explicit for clarity).
  asm volatile("s_wait_asynccnt 0x0" ::: "memory");
}

} // namespace

extern "C" void kernel_launch(void* const* d_in, const int* in_sizes, int n_in,
                              void* d_out, int out_size, void* d_ws, size_t ws_size,
                              hipStream_t stream) {
  (void)in_sizes; (void)n_in; (void)out_size; (void)d_ws; (void)ws_size;
  const float* y0    = (const float*)d_in[0];
  const float* rates = (const float*)d_in[1];
  const int*   steps = (const int*)d_in[2];
  float*       out   = (float*)d_out;

  hipLaunchKernelGGL(sidarthe_rk4_kernel, dim3(1), dim3(32), 0, stream,
                     y0, rates, steps, out);
}